// DechunkModule_2224793059971
// MI455X (gfx1250) — compile-verified
//
#include <hip/hip_runtime.h>

// ---------------------------------------------------------------------------
// EMA chunk-merge (associative scan, scalar coefficients) + token scatter.
// MI455X / gfx1250: wave32; intra-segment scan as T'(16x16) @ C(16x16) via
// V_WMMA_F32_16X16X4_F32; concept tiles staged LDS-side with
// GLOBAL_LOAD_ASYNC_TO_LDS_B128 (ASYNCcnt double buffering).
//
// Roofline: read concept 2x (128MB x2) + write out (256MB) = 512MB
// -> ~22us at 23.3 TB/s. Pure bandwidth problem; WMMA removes the length-16
// serial FMA chain per segment, async staging decouples loads from compute.
// ---------------------------------------------------------------------------

#define L_TOKS 16384
#define L_CHKS 8192
#define HIDDEN 4096
#define NBLK   64            // chunk-blocks
#define CB     128           // chunks per block
#define NSEG   (CB / 16)     // 16-chunk WMMA segments per block

typedef __attribute__((ext_vector_type(2))) float v2f;
typedef __attribute__((ext_vector_type(8))) float v8f;

#if defined(__gfx1250__) && \
    __has_builtin(__builtin_amdgcn_global_load_async_to_lds_b128) && \
    __has_builtin(__builtin_amdgcn_s_wait_asynccnt)
#define HAVE_ASYNC 1
#else
#define HAVE_ASYNC 0
#endif

// ---------------------------------------------------------------------------
// Kernel A: gather boundary indices, build p_eff[] and a[] (a[0]=0, p_eff[0]=1)
// ---------------------------------------------------------------------------
__global__ __launch_bounds__(256)
void build_coeffs(const float* __restrict__ probs,
                  const unsigned char* __restrict__ mask,
                  float* __restrict__ a_g,
                  float* __restrict__ p_g,
                  int* __restrict__ idx_g)
{
    __shared__ int cnt[256];
    const int tid  = threadIdx.x;
    const int per  = L_TOKS / 256;   // 64 tokens per thread
    const int base = tid * per;

    int c = 0;
    for (int i = 0; i < per; ++i) c += mask[base + i] ? 1 : 0;
    cnt[tid] = c;
    __syncthreads();

    if (tid == 0) {                   // serial exclusive scan over 256 counts
        int run = 0;
        for (int t = 0; t < 256; ++t) { int v = cnt[t]; cnt[t] = run; run += v; }
    }
    __syncthreads();

    int r = cnt[tid];
    for (int i = 0; i < per; ++i) {
        const int tok = base + i;
        if (mask[tok]) {
            if (r < L_CHKS) {
                idx_g[r] = tok;
                const float pv = probs[tok];
                p_g[r] = (r == 0) ? 1.0f : pv;          // b_0 = concept_0
                a_g[r] = (r == 0) ? 0.0f : (1.0f - pv); // a_0 = 0
            }
            ++r;
        }
    }
    if (tid == 0) idx_g[L_CHKS] = L_TOKS;               // sentinel span end
}

// ---------------------------------------------------------------------------
// Pass 1: per chunk-block aggregates. Bagg = local scan end (zero carry),
// Aagg = product of all a in block.
// ---------------------------------------------------------------------------
__global__ __launch_bounds__(256)
void ema_block_agg(const float* __restrict__ cpt,
                   const float* __restrict__ a_g,
                   const float* __restrict__ p_g,
                   float* __restrict__ bagg,
                   float* __restrict__ aagg)
{
    __shared__ float sa[CB];
    __shared__ float sp[CB];
    const int b = blockIdx.x, g = blockIdx.y, tid = threadIdx.x;

    if (tid < CB) { sa[tid] = a_g[b * CB + tid]; sp[tid] = p_g[b * CB + tid]; }
    __syncthreads();

    const int h = g * 256 + tid;
    const float* cp = cpt + (size_t)b * CB * HIDDEN + h;
    float y = 0.0f;
    for (int k = 0; k < CB; ++k)
        y = fmaf(sa[k], y, cp[(size_t)k * HIDDEN] * sp[k]);
    bagg[(size_t)b * HIDDEN + h] = y;

    if (g == 0 && tid == 0) {
        float q = 1.0f;
        for (int k = 0; k < CB; ++k) q *= sa[k];
        aagg[b] = q;
    }
}

// ---------------------------------------------------------------------------
// Pass 2: scan the 64 block aggregates -> carry-in vector per block.
// ---------------------------------------------------------------------------
__global__ __launch_bounds__(256)
void ema_carry_scan(const float* __restrict__ bagg,
                    const float* __restrict__ aagg,
                    float* __restrict__ carry)
{
    const int h = blockIdx.x * 256 + threadIdx.x;
    float c = 0.0f;
    for (int b = 0; b < NBLK; ++b) {
        carry[(size_t)b * HIDDEN + h] = c;
        c = fmaf(aagg[b], c, bagg[(size_t)b * HIDDEN + h]);
    }
}

// ---------------------------------------------------------------------------
// Pass 3: final scan via WMMA.
//  * T'[s][m][j] = p_j * prod_{q=j+1..m} a_q  and  P[s][m] = prod_{q=0..m} a_q
//    are block-uniform -> computed once into LDS by all 256 threads.
//  * Each wave owns a 16-col H tile; its 16x16 f32 concept tile is staged
//    into LDS with GLOBAL_LOAD_ASYNC_TO_LDS_B128 (double-buffered, ASYNCcnt).
//  * D = T' @ C via 4x V_WMMA_F32_16X16X4_F32 (A/B fed by ds_load),
//    then D += P (x) carry; scatter rows to token spans; carry' = row 15
//    via wave32 shuffle.
// Layouts (ISA 7.12.2): D VGPR r = rows {r, r+8} by lane half, col n=lane&15.
// A 16x4: VGPR0/1 = K {0,2}/{1,3} by half. B 4x16: VGPR0/1 = rows {0,2}/{1,3}.
// ---------------------------------------------------------------------------
__global__ __launch_bounds__(256)
void ema_scan_wmma(const float* __restrict__ cpt,
                   const float* __restrict__ a_g,
                   const float* __restrict__ p_g,
                   const int* __restrict__ idx_g,
                   const float* __restrict__ carry_g,
                   float* __restrict__ out)
{
    __shared__ float sa[CB];
    __shared__ float sp[CB];
    __shared__ int   sidx[CB + 1];
    __shared__ float sT[NSEG][256];      // T' = T . diag(p), row-major 16x16
    __shared__ float sPp[NSEG][16];      // carry prefix products
    __shared__ float tiles[8][2][256];   // per-wave double-buffered 16x16 tile

    const int b    = blockIdx.x;          // chunk-block
    const int g    = blockIdx.y;          // 128-wide H group
    const int tid  = threadIdx.x;
    const int wave = tid >> 5;
    const int lane = tid & 31;
    const int n    = lane & 15;           // column (B/C/D) and row (A)
    const int hi   = lane >> 4;           // lane half
    const int koff = hi << 1;             // K / B-row offset for this half
    const int rbase = hi << 3;            // D row base for this half
    const int c0   = b * CB;
    const int gcolw = g * 128 + wave * 16;  // wave's H-tile base column
    const int col   = gcolw + n;

    if (tid < CB)     { sa[tid] = a_g[c0 + tid]; sp[tid] = p_g[c0 + tid]; }
    if (tid < CB + 1) { sidx[tid] = idx_g[c0 + tid]; }
    __syncthreads();

    // Block-uniform precompute of T' and P for all segments (once per block).
    for (int e = tid; e < NSEG * 256; e += 256) {
        const int s = e >> 8, rem = e & 255;
        const int m = rem >> 4, j = rem & 15;
        float v = 0.0f;
        if (j <= m) {
            v = sp[s * 16 + j];
            for (int q = j + 1; q <= m; ++q) v *= sa[s * 16 + q];
        }
        sT[s][rem] = v;
    }
    if (tid < NSEG * 16) {
        const int s = tid >> 4, m = tid & 15;
        float v = 1.0f;
        for (int q = 0; q <= m; ++q) v *= sa[s * 16 + q];
        sPp[s][m] = v;
    }
    __syncthreads();

    // Async staging: lane copies 32B (2 x B128) of the wave's 16x16 tile.
    const int e0   = lane << 3;               // flat tile element = row*16+col
    const int trow = lane >> 1;
    const int tcol = (lane & 1) << 3;
    const float* gseg0 = cpt + (size_t)c0 * HIDDEN + gcolw;

    auto stage = [&](int s, int buf) {
        const float* gs = gseg0 + (size_t)(s * 16 + trow) * HIDDEN + tcol;
        float* ls = &tiles[wave][buf][e0];
#if HAVE_ASYNC
        typedef __attribute__((ext_vector_type(4))) int v4i_t;
        __builtin_amdgcn_global_load_async_to_lds_b128(
            (v4i_t*)gs, (v4i_t*)ls, 0, 0);
        __builtin_amdgcn_global_load_async_to_lds_b128(
            (v4i_t*)(gs + 4), (v4i_t*)(ls + 4), 0, 0);
#else
        const float4 v0 = *(const float4*)gs;
        const float4 v1 = *(const float4*)(gs + 4);
        *(float4*)ls = v0;
        *(float4*)(ls + 4) = v1;
#endif
    };

    float carry = carry_g[(size_t)b * HIDDEN + col];

    stage(0, 0);
#if !HAVE_ASYNC
    __syncthreads();
#endif

    for (int s = 0; s < NSEG; ++s) {
        const int buf = s & 1;
        if (s + 1 < NSEG) {
            stage(s + 1, buf ^ 1);
#if HAVE_ASYNC
            __builtin_amdgcn_s_wait_asynccnt(2);  // current tile resident
#else
            __syncthreads();
#endif
        } else {
#if HAVE_ASYNC
            __builtin_amdgcn_s_wait_asynccnt(0);
#else
            __syncthreads();
#endif
        }

        const float* tile = tiles[wave][buf];

        v8f acc = {0.f, 0.f, 0.f, 0.f, 0.f, 0.f, 0.f, 0.f};
#pragma unroll
        for (int t = 0; t < 4; ++t) {
            const int k1 = 4 * t + koff;
            const v2f A = *(const v2f*)&sT[s][n * 16 + k1];  // ds_load_b64
            v2f B;
            B.x = tile[(k1    ) * 16 + n];
            B.y = tile[(k1 + 1) * 16 + n];
            acc = __builtin_amdgcn_wmma_f32_16x16x4_f32(
                      false, A, false, B, (short)0, acc, false, false);
        }

        // Carry outer product: acc[r] += P[rbase+r] * carry
#pragma unroll
        for (int r = 0; r < 8; ++r)
            acc[r] = fmaf(sPp[s][rbase + r], carry, acc[r]);

        // Scatter merged states over token spans [idx[c], idx[c+1])
        const int sb = s * 16;
#pragma unroll
        for (int r = 0; r < 8; ++r) {
            const int li = sb + rbase + r;
            const int ts = sidx[li], te = sidx[li + 1];
            const float val = acc[r];
            for (int tok = ts; tok < te; ++tok)
                out[(size_t)tok * HIDDEN + col] = val;
        }

        // carry' = row 15 of D = acc[7] on lanes 16..31, column n
        carry = __shfl(acc[7], 16 + n, 32);
    }
}

// ---------------------------------------------------------------------------
// Launch. Workspace (floats): a[8192] p[8192] idx[8193] Aagg[64]
// Bagg[64*4096] carry[64*4096]  -> ~2.2 MB of d_ws.
// ---------------------------------------------------------------------------
extern "C" void kernel_launch(void* const* d_in, const int* in_sizes, int n_in,
                              void* d_out, int out_size, void* d_ws, size_t ws_size,
                              hipStream_t stream)
{
    const float*         cpt   = (const float*)d_in[0];          // [1,8192,4096]
    const float*         probs = (const float*)d_in[1];          // [1,16384,1]
    const unsigned char* mask  = (const unsigned char*)d_in[2];  // [1,16384] bool
    float* out = (float*)d_out;                                  // [1,16384,4096]

    float* w     = (float*)d_ws;
    float* a_g   = w;                       // 8192
    float* p_g   = w + 8192;                // 8192
    int*   idx_g = (int*)(w + 16384);       // 8193
    float* aagg  = w + 24592;               // 64 (padded)
    float* bagg  = w + 24672;               // 64*4096
    float* carry = bagg + (size_t)NBLK * HIDDEN;

    build_coeffs <<<1, 256, 0, stream>>>(probs, mask, a_g, p_g, idx_g);
    ema_block_agg<<<dim3(NBLK, HIDDEN / 256), 256, 0, stream>>>(cpt, a_g, p_g, bagg, aagg);
    ema_carry_scan<<<HIDDEN / 256, 256, 0, stream>>>(bagg, aagg, carry);
    ema_scan_wmma<<<dim3(NBLK, HIDDEN / 128), 256, 0, stream>>>(cpt, a_g, p_g, idx_g, carry, out);
}